// SimpleMessagePassing_352187319153
// MI455X (gfx1250) — compile-verified
//
#include <hip/hip_runtime.h>

#define N_NODES 50000
#define N_EDGES 800000
#define D 96

typedef __attribute__((ext_vector_type(2))) float v2f;
typedef __attribute__((ext_vector_type(8))) float v8f;

// ---------------------------------------------------------------------------
// Zero-fill helper (harness poisons d_out / d_ws with 0xAA before timing).
// ---------------------------------------------------------------------------
__global__ __launch_bounds__(256) void gnn_zero_kernel(float* __restrict__ p, int n) {
    int i = blockIdx.x * 256 + threadIdx.x;
    if (i < n) p[i] = 0.0f;
}

// ---------------------------------------------------------------------------
// Y = X @ W   (M=50000, K=96, N=96), full-precision fp32 WMMA.
// One wave32 per 16x16 output tile; 24 x V_WMMA_F32_16X16X4_F32 along K.
// A 16x4 layout (ISA 7.12.2): lane = 16*kHalf + m, vgpr j holds K = k0+2*kHalf+j
// B 4x16 layout:              lane = 16*kHalf + n, vgpr j holds K = k0+2*kHalf+j
// C/D 16x16:                  vgpr r, lanes 0-15 -> M=r, lanes 16-31 -> M=r+8
// ---------------------------------------------------------------------------
__global__ __launch_bounds__(32) void gnn_gemm_wmma_kernel(const float* __restrict__ X,
                                                           const float* __restrict__ W,
                                                           float* __restrict__ Y) {
    const int m0   = blockIdx.x * 16;   // 3125 tiles, exact (50000 % 16 == 0)
    const int n0   = blockIdx.y * 16;   // 6 tiles, exact
    const int lane = threadIdx.x;
    const int half = lane >> 4;         // k-half select (0 or 1)
    const int l    = lane & 15;         // m (for A) / n (for B) within tile

    const float* __restrict__ xrow = X + (size_t)(m0 + l) * D;

    v8f acc = {};
#pragma unroll
    for (int k0 = 0; k0 < D; k0 += 4) {
        const int ka = k0 + 2 * half;
        v2f a, b;
        a.x = xrow[ka];
        a.y = xrow[ka + 1];
        b.x = W[(size_t)ka * D + n0 + l];
        b.y = W[(size_t)(ka + 1) * D + n0 + l];
        // 8 args: (neg_a, A, neg_b, B, c_mod, C, reuse_a, reuse_b)
        acc = __builtin_amdgcn_wmma_f32_16x16x4_f32(false, a, false, b,
                                                    (short)0, acc, false, false);
    }

#pragma unroll
    for (int r = 0; r < 8; ++r) {
        Y[(size_t)(m0 + r + 8 * half) * D + (n0 + l)] = acc[r];
    }
}

// ---------------------------------------------------------------------------
// Scatter-add: one wave per edge, each lane owns 3 consecutive features.
// Y and out are L2-resident (19.2 MB each << 192 MB L2), so the random
// gather + f32 atomics resolve in L2's atomic units.
// ---------------------------------------------------------------------------
__global__ __launch_bounds__(256) void gnn_scatter_kernel(const int* __restrict__ src,
                                                          const int* __restrict__ dst,
                                                          const float* __restrict__ Y,
                                                          float* __restrict__ out,
                                                          float* __restrict__ deg) {
    const int e    = (int)((blockIdx.x * 256u + threadIdx.x) >> 5);
    const int lane = threadIdx.x & 31;
    if (e >= N_EDGES) return;

    const int s = src[e];
    const int d = dst[e];

    const float* __restrict__ ys = Y + (size_t)s * D + lane * 3;
    float* od = out + (size_t)d * D + lane * 3;

    const float m0 = ys[0];
    const float m1 = ys[1];
    const float m2 = ys[2];
    atomicAdd(od + 0, m0);
    atomicAdd(od + 1, m1);
    atomicAdd(od + 2, m2);
    if (lane == 0) atomicAdd(deg + d, 1.0f);
}

// ---------------------------------------------------------------------------
// out = deg > 0 ? out/deg + b : 0     (bias folds out of the mean; zero-degree
// rows are exactly 0 in the reference).
// ---------------------------------------------------------------------------
__global__ __launch_bounds__(256) void gnn_finalize_kernel(float* __restrict__ out,
                                                           const float* __restrict__ deg,
                                                           const float* __restrict__ bias) {
    const int i = blockIdx.x * 256 + threadIdx.x;
    if (i >= N_NODES * D) return;
    const int node = i / D;
    const float dg = deg[node];
    const float v  = out[i];
    out[i] = (dg > 0.0f) ? (v / dg + bias[i % D]) : 0.0f;
}

// ---------------------------------------------------------------------------
// Launch
// ---------------------------------------------------------------------------
extern "C" void kernel_launch(void* const* d_in, const int* in_sizes, int n_in,
                              void* d_out, int out_size, void* d_ws, size_t ws_size,
                              hipStream_t stream) {
    const float* x  = (const float*)d_in[0];          // [N_NODES, D]
    const int*   ei = (const int*)d_in[1];            // [2, N_EDGES]
    const float* W  = (const float*)d_in[2];          // [D, D]
    const float* b  = (const float*)d_in[3];          // [D]
    float* out = (float*)d_out;                       // [N_NODES, D]

    const int* src = ei;                              // row 0
    const int* dst = ei + N_EDGES;                    // row 1

    // Workspace layout: Y [N_NODES*D] floats, then deg [N_NODES] floats.
    float* Y   = (float*)d_ws;
    float* deg = (float*)((char*)d_ws + (size_t)N_NODES * D * sizeof(float));

    // 1) zero the accumulators
    gnn_zero_kernel<<<(N_NODES * D + 255) / 256, 256, 0, stream>>>(out, N_NODES * D);
    gnn_zero_kernel<<<(N_NODES + 255) / 256, 256, 0, stream>>>(deg, N_NODES);

    // 2) node-side projection with fp32 WMMA
    dim3 gemm_grid(N_NODES / 16, D / 16);
    gnn_gemm_wmma_kernel<<<gemm_grid, 32, 0, stream>>>(x, W, Y);

    // 3) edge scatter (one wave per edge)
    const long long scatter_threads = (long long)N_EDGES * 32;
    gnn_scatter_kernel<<<(int)((scatter_threads + 255) / 256), 256, 0, stream>>>(
        src, dst, Y, out, deg);

    // 4) mean + bias
    gnn_finalize_kernel<<<(N_NODES * D + 255) / 256, 256, 0, stream>>>(out, deg, b);
}